// NN_Model_47656957116877
// MI455X (gfx1250) — compile-verified
//
#include <hip/hip_runtime.h>
#include <math.h>

// ---------------------------------------------------------------------------
// Problem constants (from the reference)
// ---------------------------------------------------------------------------
#define LEN_M_PRI 16384
#define DATA_DIM  1023
#define DP1       1024
#define LEN_M     (LEN_M_PRI * DP1)   // 16,777,216
#define LAMB      1.0f

typedef __attribute__((ext_vector_type(2))) float v2f;
typedef __attribute__((ext_vector_type(4))) float v4f;
typedef __attribute__((ext_vector_type(8))) float v8f;

// Launch geometry: 2048 blocks x 256 threads = 16384 wave32 waves.
// LEN_M / (32 lanes * 4 floats) = 131072 chunks -> exactly 8 chunks/wave.
constexpr int BLOCKS          = 2048;
constexpr int TPB             = 256;
constexpr int WAVES_PER_BLOCK = TPB / 32;
constexpr int NWAVES          = BLOCKS * WAVES_PER_BLOCK;
constexpr int CHUNK           = 32 * 4;                 // floats per wave-iter per array
constexpr int TOTAL_CHUNKS    = LEN_M / CHUNK;          // 131072

// ---------------------------------------------------------------------------
// Bulk kernel: Sum(v) and Sum(m*m) over 2 x 16.7M floats.
// WMMA f32_16x16x4 with A = ones acts as a 64-element f32 reduction per issue:
// D[r][n] = sum_k B[k][n] + C[r][n]  (column sums, replicated over 16 rows).
// ---------------------------------------------------------------------------
__global__ __launch_bounds__(TPB) void kl_bulk(const float* __restrict__ m,
                                               const float* __restrict__ v,
                                               float* __restrict__ ws) {
    const int lane = threadIdx.x & 31;
    const int wib  = threadIdx.x >> 5;
    const int waveId = blockIdx.x * WAVES_PER_BLOCK + wib;

    v8f accV = {};   // accumulates Sum(v)
    v8f accM = {};   // accumulates Sum(m*m)
    v2f ones;
    ones.x = 1.0f; ones.y = 1.0f;

    // Wave-uniform loop: waveId/TOTAL_CHUNKS/NWAVES are lane-invariant, so
    // EXEC stays all-1s through every WMMA (ISA requirement).
    for (int c = waveId; c < TOTAL_CHUNKS; c += NWAVES) {
        const long base = (long)c * CHUNK + lane * 4;
        const v4f vv = *(const v4f*)(v + base);   // global_load_b128, coalesced
        const v4f mm = *(const v4f*)(m + base);   // global_load_b128, coalesced

        v2f b0; b0.x = vv.x;       b0.y = vv.y;
        v2f b1; b1.x = vv.z;       b1.y = vv.w;
        v2f s0; s0.x = mm.x*mm.x;  s0.y = mm.y*mm.y;
        v2f s1; s1.x = mm.z*mm.z;  s1.y = mm.w*mm.w;

        accV = __builtin_amdgcn_wmma_f32_16x16x4_f32(false, ones, false, b0,
                                                     (short)0, accV, false, false);
        accV = __builtin_amdgcn_wmma_f32_16x16x4_f32(false, ones, false, b1,
                                                     (short)0, accV, false, false);
        accM = __builtin_amdgcn_wmma_f32_16x16x4_f32(false, ones, false, s0,
                                                     (short)0, accM, false, false);
        accM = __builtin_amdgcn_wmma_f32_16x16x4_f32(false, ones, false, s1,
                                                     (short)0, accM, false, false);
    }

    // Every row of D holds identical column sums, so the sum of all 256
    // accumulator elements is 16x the true wave total. 1/16 is exact.
    float pV = 0.0f, pM = 0.0f;
#pragma unroll
    for (int i = 0; i < 8; ++i) { pV += accV[i]; pM += accM[i]; }
#pragma unroll
    for (int off = 16; off > 0; off >>= 1) {
        pV += __shfl_xor(pV, off, 32);
        pM += __shfl_xor(pM, off, 32);
    }
    pV *= (1.0f / 16.0f);
    pM *= (1.0f / 16.0f);

    __shared__ float sV[WAVES_PER_BLOCK];
    __shared__ float sM[WAVES_PER_BLOCK];
    if (lane == 0) { sV[wib] = pV; sM[wib] = pM; }
    __syncthreads();
    if (threadIdx.x == 0) {
        float tV = 0.0f, tM = 0.0f;
#pragma unroll
        for (int i = 0; i < WAVES_PER_BLOCK; ++i) { tV += sV[i]; tM += sM[i]; }
        ws[2 * blockIdx.x + 0] = tV;
        ws[2 * blockIdx.x + 1] = tM;
    }
}

// ---------------------------------------------------------------------------
// Final kernel: combine block partials + the small 16K-element terms.
// Single block, deterministic reduction order, double for the final tree.
// ---------------------------------------------------------------------------
__global__ __launch_bounds__(TPB) void kl_final(const float* __restrict__ ws,
                                                const float* __restrict__ m_pri,
                                                const float* __restrict__ v_pri,
                                                float* __restrict__ out) {
    double sV = 0.0, sM = 0.0, sVp = 0.0, sDot = 0.0, sLog = 0.0;

    for (int i = threadIdx.x; i < BLOCKS; i += TPB) {
        sV += (double)ws[2 * i + 0];
        sM += (double)ws[2 * i + 1];
    }
    for (int i = threadIdx.x; i < LEN_M_PRI; i += TPB) {
        const float vp = v_pri[i];
        const float mp = m_pri[i];
        sVp  += (double)(LAMB * vp);
        sDot += (double)(mp * mp);
        sLog += (double)logf(1.0f / (LAMB * vp));
    }

#pragma unroll
    for (int off = 16; off > 0; off >>= 1) {
        sV   += __shfl_xor(sV,   off, 32);
        sM   += __shfl_xor(sM,   off, 32);
        sVp  += __shfl_xor(sVp,  off, 32);
        sDot += __shfl_xor(sDot, off, 32);
        sLog += __shfl_xor(sLog, off, 32);
    }

    __shared__ double red[5][WAVES_PER_BLOCK];
    const int lane = threadIdx.x & 31;
    const int wib  = threadIdx.x >> 5;
    if (lane == 0) {
        red[0][wib] = sV;  red[1][wib] = sM;   red[2][wib] = sVp;
        red[3][wib] = sDot; red[4][wib] = sLog;
    }
    __syncthreads();
    if (threadIdx.x == 0) {
        double tV = 0, tM = 0, tVp = 0, tDot = 0, tLog = 0;
#pragma unroll
        for (int i = 0; i < WAVES_PER_BLOCK; ++i) {
            tV += red[0][i]; tM += red[1][i]; tVp += red[2][i];
            tDot += red[3][i]; tLog += red[4][i];
        }
        // kl1 = sum_rows 0.5*(sum_lamb_v + lamb_means_prod + sum_lamb_v - DATA_DIM)
        const double kl1 = 0.5 * (2.0 * (double)LAMB * tV + (double)LAMB * tM)
                         - 0.5 * (double)LEN_M_PRI * (double)DATA_DIM;
        const double kl2 = 0.5 * ((double)LAMB * tVp + (double)LAMB * tDot
                                  + tLog - (double)LEN_M_PRI);
        out[0] = (float)(kl1 + kl2);
    }
}

// ---------------------------------------------------------------------------
// Entry point. Inputs: d_in[0]=x (unused), d_in[1]=y (unused), d_in[2]=params.
// params layout: [m | m_pri | v | v_pri].
// ---------------------------------------------------------------------------
extern "C" void kernel_launch(void* const* d_in, const int* in_sizes, int n_in,
                              void* d_out, int out_size, void* d_ws, size_t ws_size,
                              hipStream_t stream) {
    const float* params = (const float*)d_in[2];
    const float* m      = params;
    const float* m_pri  = params + LEN_M;
    const float* v      = params + LEN_M + LEN_M_PRI;
    const float* v_pri  = params + LEN_M + LEN_M_PRI + LEN_M;  // LEN_V == LEN_M
    float* ws  = (float*)d_ws;   // 2 floats per block, fully rewritten each call
    float* out = (float*)d_out;

    kl_bulk <<<BLOCKS, TPB, 0, stream>>>(m, v, ws);
    kl_final<<<1,      TPB, 0, stream>>>(ws, m_pri, v_pri, out);
}